// Net_43447889166388
// MI455X (gfx1250) — compile-verified
//
#include <hip/hip_runtime.h>
#include <math.h>

// ---------------------------------------------------------------------------
// MI455X / gfx1250 implementation.
// FP32 WMMA (v_wmma_f32_16x16x4_f32) for all GEMMs to keep reference fp32
// semantics. Everything is L2-resident (<25MB).
//  - GEMM: 64-row x 16-col tiles per wave (4 WMMA accumulators share each
//    B fragment -> ~1.5 loads per wmma), branchless interior (clamped rows,
//    value-selected ragged-K tail). Named row pointers (not an array) so the
//    A loads stay in the GLOBAL address space (flat_load would also tie up
//    DScnt per ISA 10.1.1).
//  - Attention: F panel staged once in LDS (padded stride, bank-conflict
//    free), WMMA A-fragments come from ds_load.
// ---------------------------------------------------------------------------

typedef __attribute__((ext_vector_type(2))) float v2f;
typedef __attribute__((ext_vector_type(8))) float v8f;

__device__ __forceinline__ v8f wmma4(v2f a, v2f b, v8f c) {
  // (neg_a, A, neg_b, B, c_mod, C, reuse_a, reuse_b)
  return __builtin_amdgcn_wmma_f32_16x16x4_f32(false, a, false, b, (short)0, c,
                                               false, false);
}

// ---------------------------------------------------------------------------
// Generic row-batched GEMM:  C[r][n] = act( sum_k A[src(r)][k]*W[k][n]
//                                           + biasScale*bias[n] + add[r][n] )
// src(r) = (r / rpo)*rpi + (r % rpo)*step   (row decimation for shortcut path)
// One wave computes a 64x16 output tile (4 stacked 16x16 WMMA tiles that share
// every B fragment). N multiple of 16; K, M arbitrary.
// ---------------------------------------------------------------------------
__global__ void gemm_wmma(const float* __restrict__ A, const float* __restrict__ W,
                          const float* __restrict__ bias, const float* __restrict__ add,
                          float* __restrict__ C,
                          int M, int K, int N,
                          int rpo, int rpi, int step,
                          float biasScale, int act) {
  const int tilesN = N >> 4;
  const int tg = blockIdx.x / tilesN;       // 64-row group
  const int tn = blockIdx.x % tilesN;
  const int lane = threadIdx.x;
  const int half = lane >> 4;               // selects K sub-pair {0,1} vs {2,3}
  const int l15  = lane & 15;
  const int ncol = tn * 16 + l15;

  long srcoff[4];
#pragma unroll
  for (int t = 0; t < 4; ++t) {
    const int row = tg * 64 + t * 16 + l15;       // A row (both lane halves)
    const int rowc = row < M ? row : M - 1;       // clamp: never stored
    srcoff[t] = ((long)(rowc / rpo) * rpi + (long)(rowc % rpo) * step) * (long)K;
  }
  // named pointers (not an array) keep GLOBAL addrspace inference
  const float* ap0 = A + srcoff[0] + half * 2;
  const float* ap1 = A + srcoff[1] + half * 2;
  const float* ap2 = A + srcoff[2] + half * 2;
  const float* ap3 = A + srcoff[3] + half * 2;
  const float* wp  = W + (long)(half * 2) * N + ncol;
  const long wstep = 4L * N;

  v8f acc0 = {}, acc1 = {}, acc2 = {}, acc3 = {};
  const int Kmain = K & ~3;
  int k = 0;
  for (; k < Kmain; k += 4) {
    v2f b;
    b.x = wp[0];
    b.y = wp[N];
    v2f a0, a1, a2, a3;
    a0.x = ap0[0]; a0.y = ap0[1];
    a1.x = ap1[0]; a1.y = ap1[1];
    a2.x = ap2[0]; a2.y = ap2[1];
    a3.x = ap3[0]; a3.y = ap3[1];
    acc0 = wmma4(a0, b, acc0);
    acc1 = wmma4(a1, b, acc1);
    acc2 = wmma4(a2, b, acc2);
    acc3 = wmma4(a3, b, acc3);
    ap0 += 4; ap1 += 4; ap2 += 4; ap3 += 4;
    wp += wstep;
  }
  if (k < K) {  // ragged tail (1..3): clamp addresses, zero-select A values
    const int k0 = k + half * 2, k1 = k0 + 1;
    const int k0c = k0 < K ? k0 : K - 1;
    const int k1c = k1 < K ? k1 : K - 1;
    v2f b;
    b.x = W[(long)k0c * N + ncol];
    b.y = W[(long)k1c * N + ncol];
    {
      const float* Ar = A + srcoff[0];
      v2f a; a.x = (k0 < K) ? Ar[k0c] : 0.f; a.y = (k1 < K) ? Ar[k1c] : 0.f;
      acc0 = wmma4(a, b, acc0);
    }
    {
      const float* Ar = A + srcoff[1];
      v2f a; a.x = (k0 < K) ? Ar[k0c] : 0.f; a.y = (k1 < K) ? Ar[k1c] : 0.f;
      acc1 = wmma4(a, b, acc1);
    }
    {
      const float* Ar = A + srcoff[2];
      v2f a; a.x = (k0 < K) ? Ar[k0c] : 0.f; a.y = (k1 < K) ? Ar[k1c] : 0.f;
      acc2 = wmma4(a, b, acc2);
    }
    {
      const float* Ar = A + srcoff[3];
      v2f a; a.x = (k0 < K) ? Ar[k0c] : 0.f; a.y = (k1 < K) ? Ar[k1c] : 0.f;
      acc3 = wmma4(a, b, acc3);
    }
  }

  const float bn = biasScale * bias[ncol];
  if (tg * 64 + 64 <= M) {  // full 64-row group: wave-uniform fast path
#pragma unroll
    for (int t = 0; t < 4; ++t) {
      const v8f& accv = (t == 0) ? acc0 : (t == 1) ? acc1 : (t == 2) ? acc2 : acc3;
#pragma unroll
      for (int r = 0; r < 8; ++r) {
        const int orow = tg * 64 + t * 16 + r + half * 8;  // D: VGPR r -> M=r+8*half
        float v = accv[r] + bn;
        if (add) v += add[(long)orow * N + ncol];
        if (act == 1) v = v > 0.f ? v : 0.01f * v;
        else if (act == 2) v = fmaxf(v, 0.f);
        C[(long)orow * N + ncol] = v;
      }
    }
  } else {
#pragma unroll
    for (int t = 0; t < 4; ++t) {
      const v8f& accv = (t == 0) ? acc0 : (t == 1) ? acc1 : (t == 2) ? acc2 : acc3;
#pragma unroll
      for (int r = 0; r < 8; ++r) {
        const int orow = tg * 64 + t * 16 + r + half * 8;
        if (orow < M) {
          float v = accv[r] + bn;
          if (add) v += add[(long)orow * N + ncol];
          if (act == 1) v = v > 0.f ? v : 0.01f * v;
          else if (act == 2) v = fmaxf(v, 0.f);
          C[(long)orow * N + ncol] = v;
        }
      }
    }
  }
}

// ---------------------------------------------------------------------------
// Brute-force KNN (K=16): thread per query point; register-resident unrolled
// 16-slot insertion (neighbor order irrelevant: msg is summed over K).
// ---------------------------------------------------------------------------
__global__ void knn_kernel(const float* __restrict__ pos, int* __restrict__ nbr,
                           int B, int N, int Ns) {
  const int q = blockIdx.x * blockDim.x + threadIdx.x;
  if (q >= B * Ns) return;
  const int b = q / Ns, i = q % Ns;
  const float* pb = pos + (long)b * N * 3;
  const float qx = pb[(long)i * 12 + 0];
  const float qy = pb[(long)i * 12 + 1];
  const float qz = pb[(long)i * 12 + 2];

  float dist[16]; int idx[16];
#pragma unroll
  for (int t = 0; t < 16; ++t) { dist[t] = 3.4e38f; idx[t] = 0; }

  for (int j = 0; j < N; ++j) {
    const float dx = pb[j * 3 + 0] - qx;
    const float dy = pb[j * 3 + 1] - qy;
    const float dz = pb[j * 3 + 2] - qz;
    const float d = dx * dx + dy * dy + dz * dz;
    if (d < dist[15]) {
#pragma unroll
      for (int s = 15; s >= 1; --s) {
        if (dist[s] > d) {
          const bool fromPrev = dist[s - 1] > d;
          dist[s] = fromPrev ? dist[s - 1] : d;
          idx[s]  = fromPrev ? idx[s - 1]  : j;
        }
      }
      if (dist[0] > d) { dist[0] = d; idx[0] = j; }
    }
  }
#pragma unroll
  for (int t = 0; t < 16; ++t) nbr[(long)q * 16 + t] = idx[t];
}

// ---------------------------------------------------------------------------
// Build F[b,i,k,:] = concat( h[b, nbr, :dh],  ri @ Wl + bl ),
// ri = [rel, rel^2], rel = pos[nbr] - pos_s.  Elementwise over F.
// ---------------------------------------------------------------------------
__global__ void build_f(const float* __restrict__ h, const float* __restrict__ pos,
                        const int* __restrict__ nbr,
                        const float* __restrict__ Wl, const float* __restrict__ bl,
                        float* __restrict__ F, int B, int N, int Ns, int dh,
                        int total) {
  const int e = blockIdx.x * blockDim.x + threadIdx.x;
  if (e >= total) return;
  const int dout = 2 * dh;
  const int c = e % dout;
  const int rem = e / dout;
  const int kk = rem % 16;
  const int q = rem / 16;               // b*Ns + i
  const int b = q / Ns, i = q % Ns;
  const int j = nbr[(long)q * 16 + kk];
  if (c < dh) {
    F[e] = h[((long)b * N + j) * dh + c];
  } else {
    const int cc = c - dh;
    const float* pj = pos + ((long)b * N + j) * 3;
    const float* pi = pos + ((long)b * N + (long)i * 4) * 3;
    const float r0 = pj[0] - pi[0], r1 = pj[1] - pi[1], r2 = pj[2] - pi[2];
    F[e] = bl[cc]
         + r0 * Wl[0 * dh + cc] + r1 * Wl[1 * dh + cc] + r2 * Wl[2 * dh + cc]
         + r0 * r0 * Wl[3 * dh + cc] + r1 * r1 * Wl[4 * dh + cc]
         + r2 * r2 * Wl[5 * dh + cc];
  }
}

// ---------------------------------------------------------------------------
// Per-point attention: one workgroup (256 thr = 8 waves) per subsampled point.
//   F staged in LDS (padded stride dout+4 -> conflict-free A-row access)
//   G = F @ Wp + bp            (WMMA, A from LDS, B from L2)
//   A = softmax(G, axis=-1)    (16 row-teams of 16 threads)
//   V[p][c] = sum_k F[k][c]*A[k][c]   (second GEMM collapses by linearity:
//   agg = V @ W2 + 16*b2, done by gemm_wmma afterwards)
// ---------------------------------------------------------------------------
__global__ void attn_kernel(const float* __restrict__ Fg,
                            const float* __restrict__ Wp, const float* __restrict__ bp,
                            float* __restrict__ V, int dout) {
  __shared__ float Fs[16 * (512 + 4)];
  __shared__ float Gs[16 * (512 + 4)];
  __shared__ float red[256];
  const int S = dout + 4;               // padded LDS row stride
  const int p = blockIdx.x;
  const float* Fp = Fg + (long)p * 16 * dout;
  const int tid = threadIdx.x;
  const int wave = tid >> 5, lane = tid & 31;
  const int half = lane >> 4, l15 = lane & 15;
  const int tiles = dout >> 4;
  const long wstep = 4L * dout;

  // ---- stage F into LDS ----
#pragma unroll
  for (int r = 0; r < 16; ++r)
    for (int c = tid; c < dout; c += 256) Fs[r * S + c] = Fp[(long)r * dout + c];
  __syncthreads();

  // ---- G = F @ Wp + bp ----
  for (int tn = wave; tn < tiles; tn += 8) {
    const int ncol = tn * 16 + l15;
    const float* ap = &Fs[l15 * S + half * 2];          // LDS (ds_load)
    const float* wp = Wp + (long)(half * 2) * dout + ncol;
    v8f acc = {};
    for (int k = 0; k < dout; k += 4) {
      v2f a, b;
      a.x = ap[0];
      a.y = ap[1];
      b.x = wp[0];
      b.y = wp[dout];
      acc = wmma4(a, b, acc);
      ap += 4;
      wp += wstep;
    }
#pragma unroll
    for (int r = 0; r < 8; ++r) Gs[(r + half * 8) * S + ncol] = acc[r] + bp[ncol];
  }
  __syncthreads();

  // ---- row-wise softmax over dout, then Gs <- F * softmax(G) ----
  const int row = tid >> 4;   // 0..15 (neighbor row)
  const int tm  = tid & 15;   // team member 0..15
  float mx = -3.4e38f;
  for (int c = tm; c < dout; c += 16) mx = fmaxf(mx, Gs[row * S + c]);
  red[row * 16 + tm] = mx;
  __syncthreads();
  float rmax = red[row * 16];
#pragma unroll
  for (int t = 1; t < 16; ++t) rmax = fmaxf(rmax, red[row * 16 + t]);
  float s = 0.f;
  for (int c = tm; c < dout; c += 16) s += __expf(Gs[row * S + c] - rmax);
  __syncthreads();
  red[row * 16 + tm] = s;
  __syncthreads();
  float rsum = 0.f;
#pragma unroll
  for (int t = 0; t < 16; ++t) rsum += red[row * 16 + t];
  const float inv = 1.f / rsum;
  for (int c = tm; c < dout; c += 16) {
    const float att = __expf(Gs[row * S + c] - rmax) * inv;
    Gs[row * S + c] = Fs[row * S + c] * att;
  }
  __syncthreads();

  // ---- V[p][c] = column sum over the 16 neighbor rows ----
  for (int c = tid; c < dout; c += blockDim.x) {
    float a0 = 0.f;
#pragma unroll
    for (int r = 0; r < 16; ++r) a0 += Gs[r * S + c];
    V[(long)p * dout + c] = a0;
  }
}

// ---- small helpers ---------------------------------------------------------
__global__ void subsample_pos(const float* __restrict__ pos, float* __restrict__ ps,
                              int B, int N, int Ns) {
  const int e = blockIdx.x * blockDim.x + threadIdx.x;
  if (e >= B * Ns * 3) return;
  const int c = e % 3, i = (e / 3) % Ns, b = e / (3 * Ns);
  ps[e] = pos[((long)b * N + (long)i * 4) * 3 + c];
}

__global__ void concat_xp(const float* __restrict__ x, const float* __restrict__ ps,
                          float* __restrict__ out, int Mrows, int dx) {
  const int e = blockIdx.x * blockDim.x + threadIdx.x;
  const int dc = dx + 3;
  if (e >= Mrows * dc) return;
  const int c = e % dc, r = e / dc;
  out[e] = (c < dx) ? x[(long)r * dx + c] : ps[(long)r * 3 + (c - dx)];
}

__global__ void maxpool_pts(const float* __restrict__ gp, float* __restrict__ out,
                            int B, int P, int C) {
  const int e = blockIdx.x * blockDim.x + threadIdx.x;
  if (e >= B * C) return;
  const int b = e / C, c = e % C;
  float m = -3.4e38f;
  for (int p = 0; p < P; ++p) m = fmaxf(m, gp[((long)b * P + p) * C + c]);
  out[e] = m;
}

__global__ void head_final(const float* __restrict__ h2, const float* __restrict__ W,
                           const float* __restrict__ bias, float* __restrict__ out) {
  const int b = threadIdx.x;            // 4 batches
  if (b >= 4) return;
  float logits[10];
  float mx = -3.4e38f;
#pragma unroll
  for (int n = 0; n < 10; ++n) {
    float s = bias[n];
    for (int k = 0; k < 256; ++k) s += h2[b * 256 + k] * W[k * 10 + n];
    logits[n] = s; mx = fmaxf(mx, s);
  }
  float se = 0.f;
#pragma unroll
  for (int n = 0; n < 10; ++n) se = se + __expf(logits[n] - mx);
  const float lse = mx + __logf(se);
#pragma unroll
  for (int n = 0; n < 10; ++n) out[b * 10 + n] = logits[n] - lse;
}

// ---------------------------------------------------------------------------
// Host orchestration
// ---------------------------------------------------------------------------
static inline int ceil_div(int a, int b) { return (a + b - 1) / b; }

static void launch_gemm(hipStream_t s, const float* A, const float* W,
                        const float* bias, const float* add, float* C,
                        int M, int K, int N, int rpo, int rpi, int step,
                        float biasScale, int act) {
  const int grid = ceil_div(M, 64) * (N / 16);
  gemm_wmma<<<dim3(grid), dim3(32), 0, s>>>(A, W, bias, add, C, M, K, N,
                                            rpo, rpi, step, biasScale, act);
}

extern "C" void kernel_launch(void* const* d_in, const int* in_sizes, int n_in,
                              void* d_out, int out_size, void* d_ws, size_t ws_size,
                              hipStream_t stream) {
  (void)in_sizes; (void)n_in; (void)out_size; (void)ws_size;
  const int B = 4;
  static const int Nv[4]   = {8192, 2048, 512, 128};
  static const int dinv[4] = {3, 35, 131, 259};
  static const int dov[4]  = {32, 128, 256, 512};

  const float* pos_in = (const float*)d_in[0];

  // workspace carving (~24.4 MB of float; all buffers written before read)
  float* ws = (float*)d_ws;
  size_t off = 0;
  auto take = [&](size_t n) { float* p = ws + off; off += (n + 255) & ~(size_t)255; return p; };
  float* PA   = take(4 * 2048 * 3);
  float* PB   = take(4 * 2048 * 3);
  float* XCAT = take(4 * 2048 * 35);
  float* XOUT = take(4 * 2048 * 32);
  float* H    = take(4 * 8192 * 16);
  float* FB   = take((size_t)4 * 2048 * 16 * 32);
  float* V    = take(4 * 2048 * 32);
  float* SC   = take(4 * 2048 * 32);
  int*   NBR  = (int*)take(4 * 2048 * 16);
  float* GP   = take(4 * 32 * 1024);
  float* GPM  = take(4 * 1024);
  float* H1   = take(4 * 512);
  float* H2   = take(4 * 256);

  const float* posCur = pos_in;
  float* posNext = PA;
  const float* x_in = pos_in;

  for (int i = 0; i < 4; ++i) {
    const int N = Nv[i], Ns = N / 4, din = dinv[i], dout = dov[i], dh = dout / 2;
    const int base = 1 + i * 10;
    const float* W1  = (const float*)d_in[base + 0];
    const float* b1  = (const float*)d_in[base + 1];
    const float* Wsc = (const float*)d_in[base + 2];
    const float* bsc = (const float*)d_in[base + 3];
    const float* Wl  = (const float*)d_in[base + 4];
    const float* bl  = (const float*)d_in[base + 5];
    const float* Wp  = (const float*)d_in[base + 6];
    const float* bp  = (const float*)d_in[base + 7];
    const float* W2  = (const float*)d_in[base + 8];
    const float* b2  = (const float*)d_in[base + 9];

    // 1) KNN of decimated points among all N points
    knn_kernel<<<dim3(ceil_div(B * Ns, 256)), dim3(256), 0, stream>>>(posCur, NBR, B, N, Ns);

    // 2) h = x @ W1 + b1   (M = B*N)
    launch_gemm(stream, x_in, W1, b1, nullptr, H, B * N, din, dh, B * N, B * N, 1, 1.f, 0);

    // 3) shortcut = x[::4] @ Wsc + bsc  (row-decimated GEMM, M = B*Ns)
    launch_gemm(stream, x_in, Wsc, bsc, nullptr, SC, B * Ns, din, dout, Ns, N, 4, 1.f, 0);

    // 4) F = concat(h[nbr], ri @ Wl + bl)
    {
      const int total = B * Ns * 16 * dout;
      build_f<<<dim3(ceil_div(total, 256)), dim3(256), 0, stream>>>(
          H, posCur, NBR, Wl, bl, FB, B, N, Ns, dh, total);
    }

    // 5) V[p] = rowsum( F * softmax(F @ Wp + bp) )   (WMMA inside)
    attn_kernel<<<dim3(B * Ns), dim3(256), 0, stream>>>(FB, Wp, bp, V, dout);

    // 6) x_out = leaky_relu( V @ W2 + 16*b2 + SC )
    launch_gemm(stream, V, W2, b2, SC, XOUT, B * Ns, dout, dout,
                B * Ns, B * Ns, 1, 16.f, 1);

    // 7) pos_s and next-block input concat(x_out, pos_s)
    subsample_pos<<<dim3(ceil_div(B * Ns * 3, 256)), dim3(256), 0, stream>>>(
        posCur, posNext, B, N, Ns);
    concat_xp<<<dim3(ceil_div(B * Ns * (dout + 3), 256)), dim3(256), 0, stream>>>(
        XOUT, posNext, XCAT, B * Ns, dout);

    posCur = posNext;
    posNext = (posNext == PA) ? PB : PA;
    x_in = XCAT;
  }

  // global pool: (B*32, 515) @ (515, 1024) -> max over the 32 points
  const float* Wg  = (const float*)d_in[41];
  const float* bg  = (const float*)d_in[42];
  launch_gemm(stream, XCAT, Wg, bg, nullptr, GP, B * 32, 515, 1024,
              B * 32, B * 32, 1, 1.f, 0);
  maxpool_pts<<<dim3(ceil_div(B * 1024, 256)), dim3(256), 0, stream>>>(GP, GPM, B, 32, 1024);

  // head
  const float* Wh1 = (const float*)d_in[43]; const float* bh1 = (const float*)d_in[44];
  const float* Wh2 = (const float*)d_in[45]; const float* bh2 = (const float*)d_in[46];
  const float* Wh3 = (const float*)d_in[47]; const float* bh3 = (const float*)d_in[48];
  launch_gemm(stream, GPM, Wh1, bh1, nullptr, H1, B, 1024, 512, B, B, 1, 1.f, 2);
  launch_gemm(stream, H1, Wh2, bh2, nullptr, H2, B, 512, 256, B, B, 1, 1.f, 2);
  head_final<<<dim3(1), dim3(32), 0, stream>>>(H2, Wh3, bh3, (float*)d_out);
}